// Learn_Siminary_loss_8452495638832
// MI455X (gfx1250) — compile-verified
//
#include <hip/hip_runtime.h>
#include <math.h>

typedef __attribute__((ext_vector_type(2))) float v2f;
typedef __attribute__((ext_vector_type(8))) float v8f;

#define NROWS 768
#define DDIM  128
#define HDIM  128

// workspace layout in floats
#define WS_P    0
#define WS_Q    (NROWS*HDIM)            // 98304
#define WS_GW   (2*NROWS*HDIM)          // 196608
#define WS_C0   (WS_GW + HDIM)          // 196736
#define WS_U    (WS_C0 + 8)             // 196744
#define WS_V    (WS_U + NROWS)          // 197512
#define WS_PART (WS_V + NROWS)          // 198280

// ---------------------------------------------------------------------------
// Kernel 1: P = F @ Wl^T, Q = F @ Wr^T via V_WMMA_F32_16X16X4_F32.
// One wave per 16x16 output tile. 768 tiles total (48 M-tiles x 8 N-tiles x 2 mats).
// A frag (16x4 f32): lane<16 -> row=lane, K={4k,4k+1}; lane>=16 -> K={4k+2,4k+3}.
// B frag (4x16 f32): B[k][n] = W_sub[n][k + matsel*128]; same K split across lane halves.
// Both operands are contiguous float2 loads (8B aligned: K-base is even).
// ---------------------------------------------------------------------------
__global__ void k_gemm(const float* __restrict__ F, const float* __restrict__ W,
                       float* __restrict__ ws) {
    const int wave = threadIdx.x >> 5;
    const int lane = threadIdx.x & 31;
    const int tile = blockIdx.x * 8 + wave;            // 0..767
    const int matsel = (tile >= 384) ? 1 : 0;          // 0 = P (Wl), 1 = Q (Wr)
    const int t     = tile - matsel * 384;
    const int row0  = (t >> 3) << 4;                   // M tile * 16
    const int col0  = (t & 7) << 4;                    // N tile * 16
    const int lr    = lane & 15;
    const int hi    = lane >> 4;
    const int kq    = hi << 1;                         // 0 or 2
    const int moff  = matsel << 7;                     // column offset inside W_sub row

    const float* arow = F + (row0 + lr) * DDIM;              // features row
    const float* brow = W + (col0 + lr) * (2 * DDIM) + moff; // W_sub row (stride 256)

    v8f c = {};
#pragma unroll
    for (int kk = 0; kk < 32; ++kk) {
        const int kbase = (kk << 2) + kq;
        v2f a = *reinterpret_cast<const v2f*>(arow + kbase);
        v2f b = *reinterpret_cast<const v2f*>(brow + kbase);
        // 8 args: (neg_a, A, neg_b, B, c_mod, C, reuse_a, reuse_b)
        c = __builtin_amdgcn_wmma_f32_16x16x4_f32(false, a, false, b,
                                                  (short)0, c, false, false);
    }

    // D layout: VGPR v, lanes 0-15 -> M=v, lanes 16-31 -> M=v+8; N = lane&15.
    float* out = ws + matsel * (NROWS * HDIM);
#pragma unroll
    for (int v = 0; v < 8; ++v)
        out[(row0 + v + 8 * hi) * HDIM + col0 + lr] = c[v];
}

// ---------------------------------------------------------------------------
// Kernel 2: column stats of P and Q; fold everything into gW[h] and scalar C0.
//   var[h] = varP[h] + varQ[h]   (cross-term-free since means are removed)
//   gW[h]  = gamma[h] / sqrt(var[h]+eps) * W_meas[h]
//   C0     = sum(beta*W_meas) + b_meas - sum(P̄*gW) - sum(Q̄*gW)
// ---------------------------------------------------------------------------
__global__ void k_stats(const float* __restrict__ gamma, const float* __restrict__ beta,
                        const float* __restrict__ Wm, const float* __restrict__ bm,
                        float* __restrict__ ws) {
    __shared__ float sMean[256], sVar[256], sRed[3 * HDIM];
    const int t = threadIdx.x;
    const int matsel = t >> 7, col = t & 127;
    const float* base = ws + matsel * (NROWS * HDIM) + col;
    float s = 0.f, ss = 0.f;
    for (int r = 0; r < NROWS; ++r) { float x = base[r * HDIM]; s += x; ss += x * x; }
    const float mean = s * (1.0f / NROWS);
    const float var  = ss * (1.0f / NROWS) - mean * mean;
    sMean[t] = mean; sVar[t] = var;
    __syncthreads();
    if (t < HDIM) {
        const float vtot = sVar[t] + sVar[t + 128];
        const float g    = gamma[t] / sqrtf(vtot + 1e-5f);
        const float gw   = g * Wm[t];
        ws[WS_GW + t] = gw;
        sRed[t]           = sMean[t]       * gw;  // du
        sRed[HDIM + t]    = sMean[t + 128] * gw;  // dv
        sRed[2*HDIM + t]  = beta[t] * Wm[t];      // c
    }
    __syncthreads();
    if (t == 0) {
        float du = 0.f, dv = 0.f, cc = 0.f;
        for (int h = 0; h < HDIM; ++h) {
            du += sRed[h]; dv += sRed[HDIM + h]; cc += sRed[2*HDIM + h];
        }
        ws[WS_C0] = cc + bm[0] - du - dv;
    }
}

// ---------------------------------------------------------------------------
// Kernel 3: u[i] = dot(P_i, gW), v[j] = dot(Q_j, gW).  6 blocks x 256 threads.
// ---------------------------------------------------------------------------
__global__ void k_uv(float* __restrict__ ws) {
    __shared__ float sGW[HDIM];
    const int t = threadIdx.x;
    if (t < HDIM) sGW[t] = ws[WS_GW + t];
    __syncthreads();
    const int g = blockIdx.x * 256 + t;                 // 0..1535
    const int matsel = (g >= NROWS) ? 1 : 0;
    const int row = g - matsel * NROWS;
    const float* p = ws + matsel * (NROWS * HDIM) + row * HDIM;
    float acc = 0.f;
#pragma unroll 8
    for (int h = 0; h < HDIM; ++h) acc += p[h] * sGW[h];
    ws[(matsel ? WS_V : WS_U) + row] = acc;
}

// ---------------------------------------------------------------------------
// Kernel 4: per-row partial loss: sum_j (sigmoid(u_i + v_j + C0) - [t_i==t_j])^2
// ---------------------------------------------------------------------------
__global__ void k_pair(const int* __restrict__ target, float* __restrict__ ws) {
    __shared__ float sV[NROWS];
    __shared__ int   sT[NROWS];
    __shared__ float sred[256];
    const int t = threadIdx.x;
    const int i = blockIdx.x;
    for (int j = t; j < NROWS; j += 256) { sV[j] = ws[WS_V + j]; sT[j] = target[j]; }
    __syncthreads();
    const float ui = ws[WS_U + i] + ws[WS_C0];
    const int   ti = target[i];
    float acc = 0.f;
    for (int j = t; j < NROWS; j += 256) {
        const float x   = ui + sV[j];
        const float sgm = 1.0f / (1.0f + __expf(-x));
        const float tgt = (ti == sT[j]) ? 1.0f : 0.0f;
        const float d   = sgm - tgt;
        acc += d * d;
    }
    sred[t] = acc;
    __syncthreads();
    for (int off = 128; off > 0; off >>= 1) {
        if (t < off) sred[t] += sred[t + off];
        __syncthreads();
    }
    if (t == 0) ws[WS_PART + i] = sred[0];
}

// ---------------------------------------------------------------------------
// Kernel 5: final reduce of 768 partials -> loss (plain store, no zeroing needed).
// ---------------------------------------------------------------------------
__global__ void k_final(const float* __restrict__ ws, float* __restrict__ out) {
    __shared__ float sred[256];
    const int t = threadIdx.x;
    float acc = 0.f;
    for (int i = t; i < NROWS; i += 256) acc += ws[WS_PART + i];
    sred[t] = acc;
    __syncthreads();
    for (int off = 128; off > 0; off >>= 1) {
        if (t < off) sred[t] += sred[t + off];
        __syncthreads();
    }
    if (t == 0) out[0] = sred[0] * (1.0f / ((float)NROWS * (float)NROWS));
}

extern "C" void kernel_launch(void* const* d_in, const int* in_sizes, int n_in,
                              void* d_out, int out_size, void* d_ws, size_t ws_size,
                              hipStream_t stream) {
    const float* F     = (const float*)d_in[0];   // features (768,128)
    const float* W     = (const float*)d_in[1];   // W_sub (128,256)
    // d_in[2] = b_sub: cancels out under batch-norm, unused.
    const float* gamma = (const float*)d_in[3];   // (128)
    const float* beta  = (const float*)d_in[4];   // (128)
    const float* Wm    = (const float*)d_in[5];   // W_meas (1,128)
    const float* bm    = (const float*)d_in[6];   // b_meas (1)
    const int*   tgt   = (const int*)d_in[7];     // target (768)
    float* ws  = (float*)d_ws;
    float* out = (float*)d_out;

    k_gemm <<<96,  256, 0, stream>>>(F, W, ws);
    k_stats<<<1,   256, 0, stream>>>(gamma, beta, Wm, bm, ws);
    k_uv   <<<6,   256, 0, stream>>>(ws);
    k_pair <<<768, 256, 0, stream>>>(tgt, ws);
    k_final<<<1,   256, 0, stream>>>(ws, out);
}